// SW_VTB_44392781971918
// MI455X (gfx1250) — compile-verified
//
#include <hip/hip_runtime.h>

typedef float v2f __attribute__((ext_vector_type(2)));
typedef float v8f __attribute__((ext_vector_type(8)));

#define B_   2
#define C_   256
#define N_   110592
#define EPSV 1e-5f
#define NCHUNK_GRAM 54
#define GRAM_NPW (N_ / NCHUNK_GRAM) /* 2048 */

__device__ __forceinline__ v8f wmma_f32(v2f a, v2f b, v8f c) {
  // D = A(16x4 f32) * B(4x16 f32) + C(16x16 f32)
  return __builtin_amdgcn_wmma_f32_16x16x4_f32(false, a, false, b, (short)0, c,
                                               false, false);
}

// Async 16-byte global -> LDS transfer (per lane), tracked by ASYNCcnt.
// LDS operand = wave-relative LDS byte address = low 32 bits of generic ptr.
__device__ __forceinline__ void async_ld16(const float* g, const float* l) {
  asm volatile("global_load_async_to_lds_b128 %0, %1, off"
               :
               : "v"((unsigned)(unsigned long long)l),
                 "v"((unsigned long long)g)
               : "memory");
}

__device__ __forceinline__ void wait_async0() {
  asm volatile("s_wait_asynccnt 0x0" ::: "memory");
}

// ---------------------------------------------------------------------------
// Zero S and rowsum accumulators (they are accumulated with atomics).
// ---------------------------------------------------------------------------
__global__ void zero_kernel(float* p, int n) {
  int i = blockIdx.x * 256 + threadIdx.x;
  if (i < n) p[i] = 0.0f;
}

// ---------------------------------------------------------------------------
// Pass 1: raw Gram S[b] = x_b * x_b^T (256x256) and row sums, streamed over N.
// Double-buffered async-to-LDS fills overlap HBM traffic with WMMA.
// Grid: (54 n-chunks, 16 tiles of 64x64, 2 batches), 256 threads (8 waves).
// ---------------------------------------------------------------------------
__global__ __launch_bounds__(256) void gram_kernel(
    const float* __restrict__ x, float* __restrict__ S,
    float* __restrict__ rowsum) {
  __shared__ float xI[2][64 * 68];
  __shared__ float xJ[2][64 * 68];
  const int b = blockIdx.z;
  const int tile = blockIdx.y;
  const int ti0 = (tile >> 2) * 64;
  const int tj0 = (tile & 3) * 64;
  const long n0 = (long)blockIdx.x * GRAM_NPW;
  const int tid = threadIdx.x;
  const int wid = tid >> 5;
  const int lane = tid & 31;
  const int half = lane >> 4;
  const int l = lane & 15;
  const int t0 = wid, t1 = wid + 8;            // two 16x16 tiles per wave
  const int ti_a = t0 >> 2, tj_a = t0 & 3;
  const int ti_b = t1 >> 2, tj_b = t1 & 3;
  v8f accA = {};
  v8f accB = {};
  float rs = 0.0f;
  const float* xb = x + (size_t)b * C_ * N_;
  const int row = tid >> 4;          // 0..15
  const int col4 = (tid & 15) * 4;   // 0..60

  auto issue = [&](int buf, long nbase) {
    for (int p = 0; p < 4; ++p) {
      const int r = p * 16 + row;
      async_ld16(xb + (size_t)(ti0 + r) * N_ + nbase + col4,
                 &xI[buf][r * 68 + col4]);
      async_ld16(xb + (size_t)(tj0 + r) * N_ + nbase + col4,
                 &xJ[buf][r * 68 + col4]);
    }
  };

  const int NS = GRAM_NPW / 64;
  issue(0, n0);
  for (int s = 0; s < NS; ++s) {
    const int cur = s & 1;
    wait_async0();       // own async fills (into buf cur) complete
    __syncthreads();     // publish buf cur; everyone done reading buf cur^1
    if (s + 1 < NS) issue(cur ^ 1, n0 + (long)(s + 1) * 64);
    if ((tile & 3) == 0 && tid < 64) {   // rowsum only once per row block
      float s0 = 0.0f;
      for (int c = 0; c < 64; ++c) s0 += xI[cur][tid * 68 + c];
      rs += s0;
    }
    for (int k = 0; k < 64; k += 4) {
      const int kk = k + 2 * half;   // A: lanes<16 hold K=k,k+1; >=16 K=k+2,k+3
      v2f aA = *(const v2f*)&xI[cur][(ti_a * 16 + l) * 68 + kk];
      v2f bA = *(const v2f*)&xJ[cur][(tj_a * 16 + l) * 68 + kk];
      accA = wmma_f32(aA, bA, accA);
      v2f aB = *(const v2f*)&xI[cur][(ti_b * 16 + l) * 68 + kk];
      v2f bB = *(const v2f*)&xJ[cur][(tj_b * 16 + l) * 68 + kk];
      accB = wmma_f32(aB, bB, accB);
    }
  }

  float* Sb = S + (size_t)b * C_ * C_;
  for (int r = 0; r < 8; ++r) {
    const int m = r + 8 * half;          // C/D layout: VGPR r -> M=r / r+8
    atomicAdd(&Sb[(size_t)(ti0 + ti_a * 16 + m) * C_ + (tj0 + tj_a * 16 + l)],
              accA[r]);
    atomicAdd(&Sb[(size_t)(ti0 + ti_b * 16 + m) * C_ + (tj0 + tj_b * 16 + l)],
              accB[r]);
  }
  if ((tile & 3) == 0 && tid < 64)
    atomicAdd(&rowsum[b * C_ + ti0 + tid], rs);
}

// ---------------------------------------------------------------------------
// Per-(b,c) mean and rstd from row sums and Gram diagonal.
// ---------------------------------------------------------------------------
__global__ void stats_kernel(const float* __restrict__ S,
                             const float* __restrict__ rowsum, float* mean,
                             float* rstd) {
  int i = blockIdx.x * 256 + threadIdx.x;
  if (i >= B_ * C_) return;
  int b = i / C_, c = i % C_;
  float m = rowsum[i] / (float)N_;
  float v = S[(size_t)b * C_ * C_ + (size_t)c * C_ + c] / (float)N_ - m * m;
  mean[i] = m;
  rstd[i] = rsqrtf(v + EPSV);
}

// ---------------------------------------------------------------------------
// Normalized Gram: G[b,c,d] = rstd_c*rstd_d*(S[c,d] - N*m_c*m_d)
// ---------------------------------------------------------------------------
__global__ void gmat_kernel(const float* __restrict__ S,
                            const float* __restrict__ mean,
                            const float* __restrict__ rstd,
                            float* __restrict__ G) {
  int i = blockIdx.x * 256 + threadIdx.x;
  if (i >= B_ * C_ * C_) return;
  int b = i / (C_ * C_);
  int rem = i - b * (C_ * C_);
  int c = rem / C_, d = rem % C_;
  G[i] = rstd[b * C_ + c] * rstd[b * C_ + d] *
         (S[i] - (float)N_ * mean[b * C_ + c] * mean[b * C_ + d]);
}

// ---------------------------------------------------------------------------
// Generic batched 256x256x256 f32 WMMA GEMM: C[m,n] = sum_k A(m,k)*B(k,n)
// with arbitrary element strides (handles transposed operands).
// Grid: (16 tiles of 64x64, 1, nbatch), 256 threads.
// ---------------------------------------------------------------------------
__global__ __launch_bounds__(256) void gemm256_kernel(
    const float* __restrict__ A, long batA, int sAm, int sAk,
    const float* __restrict__ Bm, long batB, int sBk, int sBn,
    float* __restrict__ Cm, long batC) {
  const int bz = blockIdx.z;
  const float* Ab = A + batA * bz;
  const float* Bb = Bm + batB * bz;
  float* Cb = Cm + batC * bz;
  const int tid = threadIdx.x, wid = tid >> 5, lane = tid & 31;
  const int half = lane >> 4, l = lane & 15;
  const int t0 = wid, t1 = wid + 8;
  const int m0a = (blockIdx.x >> 2) * 64 + (t0 >> 2) * 16;
  const int n0a = (blockIdx.x & 3) * 64 + (t0 & 3) * 16;
  const int m0b = (blockIdx.x >> 2) * 64 + (t1 >> 2) * 16;
  const int n0b = (blockIdx.x & 3) * 64 + (t1 & 3) * 16;
  v8f accA = {}, accB = {};
  for (int k = 0; k < 256; k += 4) {
    const int kk = k + 2 * half;
    v2f a, bb;
    a.x = Ab[(m0a + l) * sAm + kk * sAk];
    a.y = Ab[(m0a + l) * sAm + (kk + 1) * sAk];
    bb.x = Bb[kk * sBk + (n0a + l) * sBn];
    bb.y = Bb[(kk + 1) * sBk + (n0a + l) * sBn];
    accA = wmma_f32(a, bb, accA);
    a.x = Ab[(m0b + l) * sAm + kk * sAk];
    a.y = Ab[(m0b + l) * sAm + (kk + 1) * sAk];
    bb.x = Bb[kk * sBk + (n0b + l) * sBn];
    bb.y = Bb[(kk + 1) * sBk + (n0b + l) * sBn];
    accB = wmma_f32(a, bb, accB);
  }
  for (int r = 0; r < 8; ++r) {
    const int m = r + 8 * half;
    Cb[(m0a + m) * C_ + n0a + l] = accA[r];
    Cb[(m0b + m) * C_ + n0b + l] = accB[r];
  }
}

// ---------------------------------------------------------------------------
// Row softmax: in = Wq*G*Wk^T; add N*bq*bk, softmax, /16, + outer/16 (in place)
// ---------------------------------------------------------------------------
__global__ __launch_bounds__(256) void softmax_kernel(
    float* __restrict__ scores, const float* __restrict__ bq,
    const float* __restrict__ bk, const float* __restrict__ outer) {
  __shared__ float red[256];
  const int b = blockIdx.x / C_, r = blockIdx.x % C_, tid = threadIdx.x;
  float* row = scores + (size_t)(b * C_ + r) * C_;
  float v = row[tid] + (float)N_ * bq[r] * bk[tid];
  red[tid] = v;
  __syncthreads();
  for (int s = 128; s > 0; s >>= 1) {
    if (tid < s) red[tid] = fmaxf(red[tid], red[tid + s]);
    __syncthreads();
  }
  const float mx = red[0];
  __syncthreads();
  const float e = __expf(v - mx);
  red[tid] = e;
  __syncthreads();
  for (int s = 128; s > 0; s >>= 1) {
    if (tid < s) red[tid] += red[tid + s];
    __syncthreads();
  }
  const float sum = red[0];
  row[tid] = e / (sum * 16.0f) + outer[r * C_ + tid] * (1.0f / 16.0f);
}

// ---------------------------------------------------------------------------
// A2[b,c,d] = (Mmat[c,d] + I)*rstd_d ; c2[b,c] = atten.bv - sum_d A2*mean_d
// ---------------------------------------------------------------------------
__global__ __launch_bounds__(256) void coeff_kernel(
    const float* __restrict__ Mmat, const float* __restrict__ atten,
    const float* __restrict__ bv, const float* __restrict__ mean,
    const float* __restrict__ rstd, float* __restrict__ A2,
    float* __restrict__ c2) {
  __shared__ float red[256];
  const int b = blockIdx.x / C_, c = blockIdx.x % C_, d = threadIdx.x;
  const size_t rowoff = (size_t)(b * C_ + c) * C_;
  float a2 = (Mmat[rowoff + d] + (c == d ? 1.0f : 0.0f)) * rstd[b * C_ + d];
  A2[rowoff + d] = a2;
  red[d] = atten[rowoff + d] * bv[d] - a2 * mean[b * C_ + d];
  __syncthreads();
  for (int s = 128; s > 0; s >>= 1) {
    if (d < s) red[d] += red[d + s];
    __syncthreads();
  }
  if (d == 0) c2[b * C_ + c] = red[0];
}

// ---------------------------------------------------------------------------
// Pass 2: out[b] = A2[b] * x[b] + c2[b]  (stream x once, write out once)
// Double-buffered async-to-LDS fills over the 4 K-chunks.
// Grid: (N/128, 4 c-tiles, 2 batches), 256 threads (8 waves, 4 tiles each).
// ---------------------------------------------------------------------------
__global__ __launch_bounds__(256) void apply_kernel(
    const float* __restrict__ x, const float* __restrict__ A2,
    const float* __restrict__ c2, float* __restrict__ out) {
  __shared__ float a2s[2][64 * 68];   // [c row][d col], padded
  __shared__ float xs[2][64 * 132];   // [d row][n col], padded
  const int b = blockIdx.z;
  const int c0 = blockIdx.y * 64;
  const long n0 = (long)blockIdx.x * 128;
  const int tid = threadIdx.x, wid = tid >> 5, lane = tid & 31;
  const int half = lane >> 4, l = lane & 15;
  const int cti = wid & 3;     // c tile 0..3
  const int njb = wid >> 2;    // n tile base 0/1; tiles njb, njb+2, njb+4, njb+6
  v8f acc[4] = {};
  const float* xb = x + (size_t)b * C_ * N_;
  const float* A2b = A2 + (size_t)b * C_ * C_;

  auto issue = [&](int buf, int kc) {
    {  // A2 block: rows c0..c0+63, cols kc*64..+63
      const int row = tid >> 4, col4 = (tid & 15) * 4;
      for (int p = 0; p < 4; ++p) {
        const int r = p * 16 + row;
        async_ld16(A2b + (size_t)(c0 + r) * C_ + kc * 64 + col4,
                   &a2s[buf][r * 68 + col4]);
      }
    }
    {  // x block: d rows kc*64..+63, n cols n0..n0+127
      const int row = tid >> 5, col4 = (tid & 31) * 4;
      for (int p = 0; p < 8; ++p) {
        const int r = p * 8 + row;
        async_ld16(xb + (size_t)(kc * 64 + r) * N_ + n0 + col4,
                   &xs[buf][r * 132 + col4]);
      }
    }
  };

  issue(0, 0);
  for (int kc = 0; kc < 4; ++kc) {
    const int cur = kc & 1;
    wait_async0();
    __syncthreads();
    if (kc + 1 < 4) issue(cur ^ 1, kc + 1);
    for (int k = 0; k < 64; k += 4) {
      const int kk = k + 2 * half;
      v2f a = *(const v2f*)&a2s[cur][(cti * 16 + l) * 68 + kk];
#pragma unroll
      for (int j = 0; j < 4; ++j) {
        const int ntj = njb + 2 * j;
        v2f bb;
        bb.x = xs[cur][kk * 132 + ntj * 16 + l];
        bb.y = xs[cur][(kk + 1) * 132 + ntj * 16 + l];
        acc[j] = wmma_f32(a, bb, acc[j]);
      }
    }
  }

  float* ob = out + (size_t)b * C_ * N_;
  for (int j = 0; j < 4; ++j) {
    const int ntj = njb + 2 * j;
    for (int r = 0; r < 8; ++r) {
      const int m = r + 8 * half;
      const int c = c0 + cti * 16 + m;
      ob[(size_t)c * N_ + n0 + ntj * 16 + l] = acc[j][r] + c2[b * C_ + c];
    }
  }
}

// ---------------------------------------------------------------------------
extern "C" void kernel_launch(void* const* d_in, const int* in_sizes, int n_in,
                              void* d_out, int out_size, void* d_ws,
                              size_t ws_size, hipStream_t stream) {
  const float* x = (const float*)d_in[0];
  const float* Wq = (const float*)d_in[1];
  const float* bq = (const float*)d_in[2];
  const float* Wk = (const float*)d_in[3];
  const float* bk = (const float*)d_in[4];
  const float* Wv = (const float*)d_in[5];
  const float* bv = (const float*)d_in[6];
  const float* outer = (const float*)d_in[7];
  float* out = (float*)d_out;
  float* ws = (float*)d_ws;

  float* S = ws;                   // 131072 floats (later reused as A2)
  float* rowsum = S + 131072;      // 512
  float* mean = rowsum + 512;      // 512
  float* rstd = mean + 512;        // 512
  float* c2 = rstd + 512;          // 512
  float* G = c2 + 512;             // 131072 (later reused: scores -> atten)
  float* T1 = G + 131072;          // 131072 (later reused: Mmat)

  // S and rowsum are contiguous -> zero both (atomically accumulated).
  zero_kernel<<<dim3((131072 + 512 + 255) / 256), 256, 0, stream>>>(
      S, 131072 + 512);
  gram_kernel<<<dim3(NCHUNK_GRAM, 16, 2), 256, 0, stream>>>(x, S, rowsum);
  stats_kernel<<<dim3(2), 256, 0, stream>>>(S, rowsum, mean, rstd);
  gmat_kernel<<<dim3(131072 / 256), 256, 0, stream>>>(S, mean, rstd, G);
  // T1 = G * Wk^T : A=G (sAm=256,sAk=1), B(k=d,n=o)=Wk[o,d] (sBk=1,sBn=256)
  gemm256_kernel<<<dim3(16, 1, 2), 256, 0, stream>>>(G, 65536L, 256, 1, Wk, 0L,
                                                     1, 256, T1, 65536L);
  // scores = Wq * T1 -> overwrite G
  gemm256_kernel<<<dim3(16, 1, 2), 256, 0, stream>>>(Wq, 0L, 256, 1, T1,
                                                     65536L, 256, 1, G, 65536L);
  softmax_kernel<<<dim3(512), 256, 0, stream>>>(G, bq, bk, outer);
  // Mmat = atten * Wv -> overwrite T1
  gemm256_kernel<<<dim3(16, 1, 2), 256, 0, stream>>>(G, 65536L, 256, 1, Wv, 0L,
                                                     256, 1, T1, 65536L);
  coeff_kernel<<<dim3(512), 256, 0, stream>>>(T1, G, bv, mean, rstd, S, c2);
  apply_kernel<<<dim3(N_ / 128, 4, 2), 256, 0, stream>>>(x, S, c2, out);
}